// TransitionDown_30545807409259
// MI455X (gfx1250) — compile-verified
//
#include <hip/hip_runtime.h>

#define B_    8
#define N_    8192
#define CIN   64
#define COUT  128
#define M_    2048          // N * 0.25
#define KNN   16
#define ROWS  (B_ * N_)     // 65536
#define Y_ELEMS ((long)B_ * M_ * COUT)   // 2,097,152

typedef float v2f __attribute__((ext_vector_type(2)));
typedef float v8f __attribute__((ext_vector_type(8)));

// ---------------------------------------------------------------------------
// Zero the BN accumulators (d_ws is poisoned, never assume zeros).
// ---------------------------------------------------------------------------
__global__ void init_accum_kernel(float* gsum, float* gssq) {
    int t = threadIdx.x;
    if (t < COUT) { gsum[t] = 0.f; gssq[t] = 0.f; }
}

// ---------------------------------------------------------------------------
// h = X(65536x64) * W^T(64x128) via V_WMMA_F32_16X16X4_F32.
// 4 waves / block, 16 rows per wave, full 128-col sweep per wave.
// Also accumulates per-channel sum / sum^2 for BatchNorm.
// ---------------------------------------------------------------------------
__global__ __launch_bounds__(128) void gemm_bn_kernel(
        const float* __restrict__ X, const float* __restrict__ W,
        float* __restrict__ H, float* __restrict__ gsum, float* __restrict__ gssq) {
    __shared__ float sW[COUT * CIN];   // 32 KB
    __shared__ float sSum[COUT];
    __shared__ float sSsq[COUT];

    const int tid = threadIdx.x;
    for (int i = tid; i < (COUT * CIN) / 4; i += 128)
        ((float4*)sW)[i] = ((const float4*)W)[i];
    if (tid < COUT) { sSum[tid] = 0.f; sSsq[tid] = 0.f; }
    __syncthreads();

    const int wave = tid >> 5;
    const int lane = tid & 31;
    const int half = lane >> 4;     // K-pair select
    const int lm   = lane & 15;     // row (A) / col (B,C)
    const long row0 = ((long)blockIdx.x * 4 + wave) * 16;

    // A fragments for all 16 K-steps: lane holds X[row0+lm][4*ks + 2*half .. +1]
    v2f a[16];
    const float* xrow = X + (row0 + lm) * CIN + 2 * half;
    #pragma unroll
    for (int ks = 0; ks < 16; ++ks)
        a[ks] = *(const v2f*)(xrow + ks * 4);

    #pragma unroll 1
    for (int t = 0; t < COUT / 16; ++t) {
        const int o0 = t * 16;
        v8f acc = {};
        const float* wrow = sW + (o0 + lm) * CIN + 2 * half;  // B[k][n] = W[o0+n][k]
        #pragma unroll
        for (int ks = 0; ks < 16; ++ks) {
            v2f bf = *(const v2f*)(wrow + ks * 4);
            acc = __builtin_amdgcn_wmma_f32_16x16x4_f32(
                    false, a[ks], false, bf, (short)0, acc, false, false);
        }
        // C/D layout: VGPR v = row (row0 + 8*half + v), col o0+lm
        const long rbase = row0 + (half ? 8 : 0);
        float s = 0.f, q = 0.f;
        #pragma unroll
        for (int v = 0; v < 8; ++v) {
            float val = acc[v];
            H[(rbase + v) * COUT + o0 + lm] = val;
            s += val; q += val * val;
        }
        atomicAdd(&sSum[o0 + lm], s);
        atomicAdd(&sSsq[o0 + lm], q);
    }
    __syncthreads();
    if (tid < COUT) {
        atomicAdd(&gsum[tid], sSum[tid]);
        atomicAdd(&gssq[tid], sSsq[tid]);
    }
}

// ---------------------------------------------------------------------------
// Fold BN stats + gamma/beta into per-channel scale/bias.
// ---------------------------------------------------------------------------
__global__ void bn_finalize_kernel(const float* gsum, const float* gssq,
                                   const float* gamma, const float* beta,
                                   float* scale, float* bias) {
    int o = threadIdx.x;
    if (o < COUT) {
        const float inv  = 1.0f / (float)ROWS;
        float mean = gsum[o] * inv;
        float var  = gssq[o] * inv - mean * mean;
        float sc   = gamma[o] * rsqrtf(var + 1e-5f);
        scale[o] = sc;
        bias[o]  = beta[o] - mean * sc;
    }
}

// ---------------------------------------------------------------------------
// Furthest point sampling: 1 block / batch, points in dynamic LDS (96 KB),
// per-thread min-distance kept in registers (8 each), block-wide argmax.
// Writes p2 (second output) directly.
// ---------------------------------------------------------------------------
__global__ __launch_bounds__(1024) void fps_kernel(const float* __restrict__ P1,
                                                   float* __restrict__ p2out) {
    extern __shared__ float sp[];               // 3*N floats
    __shared__ float redD[1024];
    __shared__ int   redI[1024];
    float* px = sp; float* py = sp + N_; float* pz = sp + 2 * N_;

    const int b = blockIdx.x, tid = threadIdx.x;
    const float* pb = P1 + (long)b * N_ * 3;
    for (int i = tid; i < N_; i += 1024) {
        px[i] = pb[i * 3 + 0]; py[i] = pb[i * 3 + 1]; pz[i] = pb[i * 3 + 2];
    }
    __syncthreads();

    float mind[8];
    #pragma unroll
    for (int j = 0; j < 8; ++j) mind[j] = 1e10f;

    int last = 0;
    if (tid == 0) {
        p2out[((long)b * M_) * 3 + 0] = px[0];
        p2out[((long)b * M_) * 3 + 1] = py[0];
        p2out[((long)b * M_) * 3 + 2] = pz[0];
    }
    for (int m = 1; m < M_; ++m) {
        const float lx = px[last], ly = py[last], lz = pz[last];
        float bestD = -1.f; int bestI = 0;
        #pragma unroll
        for (int j = 0; j < 8; ++j) {
            const int i = tid + j * 1024;
            float dx = px[i] - lx, dy = py[i] - ly, dz = pz[i] - lz;
            float d  = dx * dx + dy * dy + dz * dz;
            float mn = fminf(mind[j], d);
            mind[j]  = mn;
            if (mn > bestD) { bestD = mn; bestI = i; }   // ascending i: keeps first max
        }
        redD[tid] = bestD; redI[tid] = bestI;
        __syncthreads();
        for (int s = 512; s > 0; s >>= 1) {
            if (tid < s) {
                float dA = redD[tid], dB = redD[tid + s];
                int   iA = redI[tid], iB = redI[tid + s];
                if (dB > dA || (dB == dA && iB < iA)) { redD[tid] = dB; redI[tid] = iB; }
            }
            __syncthreads();
        }
        last = redI[0];
        __syncthreads();
        if (tid == 0) {
            long o = ((long)b * M_ + m) * 3;
            p2out[o + 0] = px[last]; p2out[o + 1] = py[last]; p2out[o + 2] = pz[last];
        }
    }
}

// ---------------------------------------------------------------------------
// kNN (K=16 smallest sq-distances): 1 thread / query, p1 staged in LDS,
// register-resident worst-replacement top-K.
// ---------------------------------------------------------------------------
__global__ __launch_bounds__(256) void knn_kernel(const float* __restrict__ P1,
                                                  const float* __restrict__ p2out,
                                                  int* __restrict__ nbr) {
    extern __shared__ float sp[];
    float* px = sp; float* py = sp + N_; float* pz = sp + 2 * N_;
    const int b     = blockIdx.x / (M_ / 256);
    const int chunk = blockIdx.x % (M_ / 256);
    const int tid   = threadIdx.x;
    const float* pb = P1 + (long)b * N_ * 3;
    for (int i = tid; i < N_; i += 256) {
        px[i] = pb[i * 3]; py[i] = pb[i * 3 + 1]; pz[i] = pb[i * 3 + 2];
    }
    __syncthreads();

    const int m = chunk * 256 + tid;
    const long qo = ((long)b * M_ + m) * 3;
    const float qx = p2out[qo], qy = p2out[qo + 1], qz = p2out[qo + 2];

    float dist[KNN]; int idxr[KNN];
    #pragma unroll
    for (int k = 0; k < KNN; ++k) { dist[k] = 3.4e38f; idxr[k] = 0; }
    float worst = 3.4e38f; int wslot = 0;

    for (int n = 0; n < N_; ++n) {
        float dx = px[n] - qx, dy = py[n] - qy, dz = pz[n] - qz;
        float d  = dx * dx + dy * dy + dz * dz;
        if (d < worst) {
            #pragma unroll
            for (int k = 0; k < KNN; ++k)
                if (k == wslot) { dist[k] = d; idxr[k] = n; }
            worst = -1.f;
            #pragma unroll
            for (int k = 0; k < KNN; ++k)
                if (dist[k] > worst) { worst = dist[k]; wslot = k; }
        }
    }
    int* o = nbr + ((long)b * M_ + m) * KNN;
    #pragma unroll
    for (int k = 0; k < KNN; ++k) o[k] = idxr[k];
}

// ---------------------------------------------------------------------------
// y[b,m,o] = max_k relu(h[b, nbr[b,m,k], o]*scale[o] + bias[o])
// h lives in L2 (33.5 MB << 192 MB); consecutive-o threads coalesce gathers.
// ---------------------------------------------------------------------------
__global__ __launch_bounds__(256) void gather_max_kernel(
        const float* __restrict__ H, const int* __restrict__ nbr,
        const float* __restrict__ scale, const float* __restrict__ bias,
        float* __restrict__ Y) {
    const long gid = (long)blockIdx.x * 256 + threadIdx.x;
    if (gid >= Y_ELEMS) return;
    const int  o  = (int)(gid & (COUT - 1));
    const long bm = gid >> 7;
    const long b  = bm >> 11;            // M_ = 2048
    const int* nn = nbr + bm * KNN;
    const float sc = scale[o], bi = bias[o];
    float mx = 0.f;                      // relu output is >= 0
    #pragma unroll
    for (int k = 0; k < KNN; ++k) {
        long n = nn[k];
        float v = fmaxf(H[(b * N_ + n) * COUT + o] * sc + bi, 0.f);
        mx = fmaxf(mx, v);
    }
    Y[gid] = mx;
}

// ---------------------------------------------------------------------------
extern "C" void kernel_launch(void* const* d_in, const int* in_sizes, int n_in,
                              void* d_out, int out_size, void* d_ws, size_t ws_size,
                              hipStream_t stream) {
    const float* x     = (const float*)d_in[0];
    const float* p1    = (const float*)d_in[1];
    const float* W     = (const float*)d_in[2];
    const float* gamma = (const float*)d_in[3];
    const float* beta  = (const float*)d_in[4];

    float* y  = (float*)d_out;
    float* p2 = y + Y_ELEMS;

    // workspace layout
    float* H     = (float*)d_ws;                 // 65536*128 floats (33.5 MB)
    float* gsum  = H + (long)ROWS * COUT;        // 128
    float* gssq  = gsum + COUT;                  // 128
    float* scale = gssq + COUT;                  // 128
    float* bias  = scale + COUT;                 // 128
    int*   nbr   = (int*)(bias + COUT);          // B*M*K ints

    const size_t ldsPts = (size_t)3 * N_ * sizeof(float);   // 96 KB

    init_accum_kernel<<<1, 256, 0, stream>>>(gsum, gssq);
    gemm_bn_kernel<<<ROWS / 64, 128, 0, stream>>>(x, W, H, gsum, gssq);
    bn_finalize_kernel<<<1, 128, 0, stream>>>(gsum, gssq, gamma, beta, scale, bias);
    fps_kernel<<<B_, 1024, ldsPts, stream>>>(p1, p2);
    knn_kernel<<<B_ * (M_ / 256), 256, ldsPts, stream>>>(p1, p2, nbr);
    gather_max_kernel<<<(unsigned)((Y_ELEMS + 255) / 256), 256, 0, stream>>>(
        H, nbr, scale, bias, y);
}